// _ProposalLayer_54631984005138
// MI455X (gfx1250) — compile-verified
//
#include <hip/hip_runtime.h>
#include <stdint.h>

// ---------------------------------------------------------------------------
// ProposalLayer for MI455X (gfx1250, wave32).
// B=8, A=9 anchors, H=W=128, stride 16. PRE_NMS=2000, POST_NMS=300, thr=0.7.
// ---------------------------------------------------------------------------

#define NANCH  9
#define HDIM   128
#define WDIM   128
#define HW     16384              // 128*128
#define NPB    (NANCH * HW)       // 147456 candidates per batch
#define NBATCH 8
#define TOPK   2000
#define PADK   2048               // power-of-two pad for bitonic sort / LDS tile
#define POSTK  300
#define SUPW   32                 // u64 words per suppression row (32*64 = 2048 bits)

typedef __attribute__((ext_vector_type(4))) unsigned int uint4v;
typedef __attribute__((ext_vector_type(8))) unsigned int uint8v;

// Anchor widths/heights (x2-x1+1, y2-y1+1) from generate_anchors(); all anchor
// centers are exactly (8,8) before the feature-stride shift.
__constant__ float c_wa[NANCH] = {184.f, 368.f, 736.f, 128.f, 256.f, 512.f,  88.f, 176.f, 352.f};
__constant__ float c_ha[NANCH] = { 96.f, 192.f, 384.f, 128.f, 256.f, 512.f, 176.f, 352.f, 704.f};

__device__ __forceinline__ unsigned f2key(float f) {
  // monotonic float -> uint mapping (larger key == larger float)
  unsigned u = __float_as_uint(f);
  return u ^ ((unsigned)(((int)u) >> 31) | 0x80000000u);
}

__device__ __forceinline__ unsigned long long bcast64(unsigned long long v, int srcLane) {
  int lo = __shfl((int)(unsigned)(v & 0xFFFFFFFFull), srcLane, 32);
  int hi = __shfl((int)(unsigned)(v >> 32), srcLane, 32);
  return ((unsigned long long)(unsigned)hi << 32) | (unsigned long long)(unsigned)lo;
}

// ---------------------------------------------------------------------------
// Kernel 1: per-batch radix-select of the 2000th-largest objectness key, then
// compaction of all candidates >= threshold into u64 (key<<32 | ~refIdx).
// Objectness scores for batch b are channels [A, 2A) -> a contiguous slab of
// NPB floats at scores + b*2*A*HW + A*HW, indexed i = a*HW + pos.
// Reference flatten order is refIdx = pos*A + a (tie-break: lower refIdx wins).
// ---------------------------------------------------------------------------
__global__ __launch_bounds__(1024) void topk_select_kernel(
    const float* __restrict__ scores, unsigned long long* __restrict__ cand) {
  const int b = blockIdx.x;
  const int tid = threadIdx.x;
  const int nt = blockDim.x;
  const float* s = scores + (size_t)b * 2 * NANCH * HW + (size_t)NANCH * HW;

  __shared__ unsigned int hist[256];
  __shared__ unsigned int sh_prefix;
  __shared__ int sh_k;
  __shared__ unsigned int sh_cnt;

  if (tid == 0) { sh_prefix = 0u; sh_k = TOPK; sh_cnt = 0u; }
  __syncthreads();

  for (int shift = 24; shift >= 0; shift -= 8) {
    for (int i = tid; i < 256; i += nt) hist[i] = 0u;
    __syncthreads();
    const unsigned pref = sh_prefix;
    const unsigned maskHigh = (shift == 24) ? 0u : (0xFFFFFFFFu << (shift + 8));
    for (int i = tid; i < NPB; i += nt) {
      unsigned key = f2key(s[i]);
      if ((key & maskHigh) == pref)
        atomicAdd(&hist[(key >> shift) & 255u], 1u);
    }
    __syncthreads();
    if (tid == 0) {
      int k = sh_k;
      unsigned cum = 0u;
      for (int bin = 255; bin >= 0; --bin) {
        unsigned c = hist[bin];
        if (cum + c >= (unsigned)k) {
          sh_prefix = pref | ((unsigned)bin << shift);
          sh_k = k - (int)cum;
          break;
        }
        cum += c;
      }
    }
    __syncthreads();
  }

  const unsigned T = sh_prefix;  // exact key of the 2000th largest
  unsigned long long* outC = cand + (size_t)b * PADK;
  for (int i = tid; i < NPB; i += nt) {
    unsigned key = f2key(s[i]);
    if (key >= T) {
      unsigned p = atomicAdd(&sh_cnt, 1u);
      if (p < PADK) {
        unsigned refIdx = (unsigned)((i & (HW - 1)) * NANCH + (i >> 14));
        outC[p] = ((unsigned long long)key << 32) |
                  (unsigned long long)(0xFFFFFFFFu - refIdx);
      }
    }
  }
  __syncthreads();
  unsigned c = sh_cnt; if (c > PADK) c = PADK;
  for (unsigned i = c + tid; i < PADK; i += nt) outC[i] = 0ull;
}

// ---------------------------------------------------------------------------
// Kernel 2: bitonic sort (descending) of the 2048 candidates in LDS, then
// decode+clip proposal boxes ONLY for the 2000 winners (avoids materializing
// all 1.18M boxes -> saves ~38 MB of HBM traffic).
// The 16 KB key tile is staged via gfx1250 async-LDS loads (ASYNCcnt path).
// ---------------------------------------------------------------------------
__global__ __launch_bounds__(1024) void sort_decode_kernel(
    const unsigned long long* __restrict__ cand,
    const float* __restrict__ deltas,
    const float* __restrict__ im_info,
    float4* __restrict__ topBoxes) {
  const int b = blockIdx.x;
  const int tid = threadIdx.x;
  const int nt = blockDim.x;

  __shared__ unsigned long long sh[PADK];  // 16 KB

  // Async stage: per-lane 8B copies, memory -> LDS, no VGPR round trip.
  {
    const unsigned long long* src = cand + (size_t)b * PADK;
    for (int e = tid; e < PADK; e += nt) {
      unsigned ldsAddr = (unsigned)(uintptr_t)(&sh[e]);  // low 32 bits = LDS offset
      unsigned long long gAddr = (unsigned long long)(uintptr_t)(src + e);
      asm volatile("global_load_async_to_lds_b64 %0, %1, off"
                   :: "v"(ldsAddr), "v"(gAddr) : "memory");
    }
    asm volatile("s_wait_asynccnt 0" ::: "memory");
  }
  __syncthreads();

  for (int k = 2; k <= PADK; k <<= 1) {
    for (int j = k >> 1; j > 0; j >>= 1) {
      for (int i = tid; i < PADK; i += nt) {
        int ixj = i ^ j;
        if (ixj > i) {
          unsigned long long va = sh[i], vb = sh[ixj];
          bool descSeg = ((i & k) == 0);
          bool doSwap = descSeg ? (va < vb) : (va > vb);
          if (doSwap) { sh[i] = vb; sh[ixj] = va; }
        }
      }
      __syncthreads();
    }
  }

  const float maxX = im_info[b * 3 + 1] - 1.0f;
  const float maxY = im_info[b * 3 + 0] - 1.0f;
  const float* dB = deltas + (size_t)b * 4 * NANCH * HW;

  for (int jj = tid; jj < TOPK; jj += nt) {
    unsigned long long v = sh[jj];
    unsigned refIdx = 0xFFFFFFFFu - (unsigned)(v & 0xFFFFFFFFull);
    int a = (int)(refIdx % NANCH);
    int pos = (int)(refIdx / NANCH);
    int h = pos >> 7, w = pos & (WDIM - 1);
    const float* dp = dB + ((size_t)(4 * a) << 14) + pos;
    float dx = dp[0], dy = dp[HW], dw = dp[2 * HW], dh = dp[3 * HW];
    float wa = c_wa[a], ha = c_ha[a];
    float cx = 8.0f + 16.0f * (float)w + dx * wa;
    float cy = 8.0f + 16.0f * (float)h + dy * ha;
    float pw = expf(dw) * wa;
    float ph = expf(dh) * ha;
    float x1 = cx - 0.5f * pw, y1 = cy - 0.5f * ph;
    float x2 = cx + 0.5f * pw, y2 = cy + 0.5f * ph;
    x1 = fminf(fmaxf(x1, 0.f), maxX);
    y1 = fminf(fmaxf(y1, 0.f), maxY);
    x2 = fminf(fmaxf(x2, 0.f), maxX);
    y2 = fminf(fmaxf(y2, 0.f), maxY);
    topBoxes[(size_t)b * PADK + jj] = make_float4(x1, y1, x2, y2);
  }
}

// ---------------------------------------------------------------------------
// Kernel 3: upper-triangular IoU>0.7 suppression bitmatrix.
// The full 32 KB candidate-box tile is DMA'd into LDS with ONE Tensor Data
// Mover op (tensor_load_to_lds, TENSORcnt-tracked), issued by wave 0 with a
// hand-built D# descriptor:
//   group0: count=1, lds_addr, global_addr[56:0], type=2
//   group1: mask=0, data_size=8B, tile 4096x1, tensor_dim0=stride=4096
// Then each thread owns one row and emits a 2048-bit mask (32 u64 words).
// ---------------------------------------------------------------------------
#define NMSB 128
__global__ __launch_bounds__(NMSB) void nms_mask_kernel(
    const float4* __restrict__ topBoxes, unsigned long long* __restrict__ sup) {
  const int b = blockIdx.y;
  const int row = blockIdx.x * NMSB + threadIdx.x;

  __shared__ float4 shB[PADK];  // 32 KB candidate boxes

  if (threadIdx.x < 32) {  // wave 0 issues the TDM op (EXEC ignored by TDM)
    unsigned ldsBase = (unsigned)(uintptr_t)(&shB[0]);
    unsigned long long gaddr =
        (unsigned long long)(uintptr_t)(topBoxes + (size_t)b * PADK);

    uint4v g0;
    g0.x = 1u;                                               // count=1, user desc
    g0.y = ldsBase;                                          // lds_addr
    g0.z = (unsigned)(gaddr & 0xFFFFFFFFull);                // global_addr[31:0]
    g0.w = (unsigned)((gaddr >> 32) & 0x01FFFFFFull)         // global_addr[56:32]
         | (2u << 30);                                       // type=2 (image)

    uint8v g1;
    g1.s0 = 0x00030000u;   // workgroup_mask=0, data_size=3 (8B), no flags
    g1.s1 = 0x10000000u;   // atomic_barrier_addr=0 | tensor_dim0[15:0]=4096 <<16
    g1.s2 = 0x00010000u;   // tensor_dim0[31:16]=0 | tensor_dim1[15:0]=1 <<16
    g1.s3 = 0x10000000u;   // tensor_dim1[31:16]=0 | tile_dim0=4096 <<16
    g1.s4 = 0x00000001u;   // tile_dim1=1 | tile_dim2=0
    g1.s5 = 4096u;         // tensor_dim0_stride[31:0]
    g1.s6 = 0u;            // tensor_dim0_stride[47:32] | tensor_dim1_stride[15:0]
    g1.s7 = 0u;            // tensor_dim1_stride[47:16]

    asm volatile("tensor_load_to_lds %0, %1" :: "s"(g0), "s"(g1) : "memory");
    __builtin_amdgcn_s_wait_tensorcnt(0);
  }
  __syncthreads();

  if (row >= TOPK) return;
  float4 bi = shB[row];
  float areaI = (bi.z - bi.x) * (bi.w - bi.y);
  unsigned long long* outRow = sup + ((size_t)b * TOPK + row) * SUPW;

  for (int wgrp = 0; wgrp < SUPW; ++wgrp) {
    unsigned long long bits = 0ull;
    int jbase = wgrp << 6;
#pragma unroll 8
    for (int jj = 0; jj < 64; ++jj) {
      int j = jbase + jj;
      if (j > row && j < TOPK) {
        float4 bj = shB[j];  // same j across the wave -> LDS broadcast
        float xx1 = fmaxf(bi.x, bj.x), yy1 = fmaxf(bi.y, bj.y);
        float xx2 = fminf(bi.z, bj.z), yy2 = fminf(bi.w, bj.w);
        float iw = fmaxf(xx2 - xx1, 0.f), ih = fmaxf(yy2 - yy1, 0.f);
        float inter = iw * ih;
        float areaJ = (bj.z - bj.x) * (bj.w - bj.y);
        float iou = inter / (areaI + areaJ - inter + 1e-9f);
        if (iou > 0.7f) bits |= (1ull << jj);
      }
    }
    outRow[wgrp] = bits;
  }
}

// ---------------------------------------------------------------------------
// Kernel 4: single-wave greedy NMS scan + output scatter. One wave32 per batch;
// lane t owns keep bits [64t, 64t+63] in a register. Keep-bit broadcast via
// shuffles, next suppression row prefetched unconditionally to hide latency.
// Output rows are pre-initialized to [b,0,0,0,0] (matches `* valid` zeroing).
// ---------------------------------------------------------------------------
__global__ __launch_bounds__(32) void nms_reduce_kernel(
    const unsigned long long* __restrict__ sup,
    const float4* __restrict__ topBoxes,
    float* __restrict__ out) {
  const int b = blockIdx.x;
  const int lane = threadIdx.x;
  const unsigned long long* supB = sup + (size_t)b * TOPK * SUPW;
  float* outB = out + (size_t)b * POSTK * 5;

  // init all 300 rows to [b, 0,0,0,0]
  for (int r = lane; r < POSTK; r += 32) {
    outB[r * 5 + 0] = (float)b;
    outB[r * 5 + 1] = 0.f;
    outB[r * 5 + 2] = 0.f;
    outB[r * 5 + 3] = 0.f;
    outB[r * 5 + 4] = 0.f;
  }
  asm volatile("s_wait_storecnt 0" ::: "memory");

  unsigned long long keep = ~0ull;           // lane owns bits [64*lane, 64*lane+63]
  unsigned long long nxt = supB[lane];       // prefetch row 0
  for (int i = 0; i < TOPK; ++i) {
    unsigned long long cur = nxt;
    if (i + 1 < TOPK) nxt = supB[(size_t)(i + 1) * SUPW + lane];
    unsigned long long kw = bcast64(keep, i >> 6);
    if ((kw >> (i & 63)) & 1ull) keep &= ~cur;   // sup[i][j]=0 for j<=i
  }

  // mask off padding bits >= TOPK (lane 31 covers bits 1984..2047 -> 16 valid)
  unsigned long long vmask = (lane == 31) ? ((1ull << 16) - 1ull) : ~0ull;
  unsigned long long kv = keep & vmask;

  int pc = __popcll(kv);
  int excl = 0;
  for (int l = 0; l < 32; ++l) {
    int vv = __shfl(pc, l, 32);
    if (l < lane) excl += vv;
  }

  int r = excl;
  for (int bb = 0; bb < 64; ++bb) {
    if ((kv >> bb) & 1ull) {
      if (r < POSTK) {
        int i = (lane << 6) + bb;
        float4 bx = topBoxes[(size_t)b * PADK + i];
        outB[r * 5 + 1] = bx.x;
        outB[r * 5 + 2] = bx.y;
        outB[r * 5 + 3] = bx.z;
        outB[r * 5 + 4] = bx.w;
      }
      ++r;
    }
  }
}

// ---------------------------------------------------------------------------
// Host launcher. Workspace layout (all 256B-aligned, total ~4.4 MB):
//   cand     : NBATCH * PADK * 8  = 131,072 B
//   topBoxes : NBATCH * PADK * 16 = 262,144 B
//   sup      : NBATCH * TOPK * SUPW * 8 = 4,096,000 B
// ---------------------------------------------------------------------------
extern "C" void kernel_launch(void* const* d_in, const int* in_sizes, int n_in,
                              void* d_out, int out_size, void* d_ws, size_t ws_size,
                              hipStream_t stream) {
  (void)in_sizes; (void)n_in; (void)out_size; (void)ws_size;
  const float* scores = (const float*)d_in[0];
  const float* deltas = (const float*)d_in[1];
  const float* iminfo = (const float*)d_in[2];
  float* out = (float*)d_out;

  char* ws = (char*)d_ws;
  unsigned long long* cand = (unsigned long long*)(ws);
  float4* topBoxes = (float4*)(ws + 131072);
  unsigned long long* sup = (unsigned long long*)(ws + 131072 + 262144);

  topk_select_kernel<<<NBATCH, 1024, 0, stream>>>(scores, cand);
  sort_decode_kernel<<<NBATCH, 1024, 0, stream>>>(cand, deltas, iminfo, topBoxes);
  nms_mask_kernel<<<dim3((TOPK + NMSB - 1) / NMSB, NBATCH), NMSB, 0, stream>>>(topBoxes, sup);
  nms_reduce_kernel<<<NBATCH, 32, 0, stream>>>(sup, topBoxes, out);
}